// MahalalobisL2_6485400617248
// MI455X (gfx1250) — compile-verified
//
#include <hip/hip_runtime.h>
#include <cfloat>
#include <cmath>

// Problem constants (match reference)
#define B_N 2048
#define D_N 1024
#define EPS 0.05f
#define INV_B (1.0f / 2048.0f)
#define N_ITER 100

typedef __attribute__((ext_vector_type(2))) float v2f;
typedef __attribute__((ext_vector_type(8))) float v8f;

// ---------------------------------------------------------------------------
// Kernel 0: w = diag(nan_to_num(L)); init u = 1/B; zero control scalars
// ---------------------------------------------------------------------------
__global__ __launch_bounds__(1024) void prep_kernel(
    const float* __restrict__ L, float* __restrict__ w, float* __restrict__ u,
    unsigned int* __restrict__ maxbits, unsigned int* __restrict__ counter) {
  int gid = blockIdx.x * 1024 + threadIdx.x;
  if (gid < D_N) {
    float val = L[(size_t)gid * D_N + gid];
    if (val != val) val = 0.0f;                 // nan -> 0
    if (val > FLT_MAX) val = FLT_MAX;           // +inf clamp
    if (val < -FLT_MAX) val = -FLT_MAX;         // -inf clamp
    w[gid] = val;
  }
  if (gid < B_N) u[gid] = INV_B;
  if (gid == 0) { *maxbits = 0u; *counter = 0u; }
}

// ---------------------------------------------------------------------------
// Kernel 1: per-row weighted norms + weighted text matrix
//   aw[r] = sum_d audio[r,d]^2 w[d], tw[r] = sum_d text[r,d]^2 w[d]
//   Tw[r,d] = text[r,d] * w[d]
// ---------------------------------------------------------------------------
__global__ __launch_bounds__(256) void rowprep_kernel(
    const float* __restrict__ audio, const float* __restrict__ text,
    const float* __restrict__ w, float* __restrict__ Tw,
    float* __restrict__ awv, float* __restrict__ twv) {
  const int r = blockIdx.x;
  const int tid = threadIdx.x;
  float sa = 0.0f, st = 0.0f;
  for (int d = tid; d < D_N; d += 256) {
    float wd = w[d];
    float av = audio[(size_t)r * D_N + d];
    float tv = text[(size_t)r * D_N + d];
    sa += av * av * wd;
    st += tv * tv * wd;
    Tw[(size_t)r * D_N + d] = tv * wd;
  }
  __shared__ float sha[256], sht[256];
  sha[tid] = sa; sht[tid] = st;
  __syncthreads();
  for (int off = 128; off > 0; off >>= 1) {
    if (tid < off) { sha[tid] += sha[tid + off]; sht[tid] += sht[tid + off]; }
    __syncthreads();
  }
  if (tid == 0) { awv[r] = sha[0]; twv[r] = sht[0]; }
}

// ---------------------------------------------------------------------------
// Kernel 2: cross GEMM via f32 WMMA + fused distance epilogue + global max
//   Each wave: 32x32 output tile = 2x2 grid of v_wmma_f32_16x16x4_f32 accs.
//   Mraw[i,j] = sqrt(max(tw[i] + aw[j] - 2*sum_d Tw[i,d]*Au[j,d], 0))
// ---------------------------------------------------------------------------
__global__ __launch_bounds__(256) void cross_dist_kernel(
    const float* __restrict__ Au, const float* __restrict__ Tw,
    const float* __restrict__ awv, const float* __restrict__ twv,
    float* __restrict__ Mraw, unsigned int* __restrict__ maxbits) {
  const int tid  = threadIdx.x;
  const int lane = tid & 31;
  const int wave = blockIdx.x * 8 + (tid >> 5);   // 4096 waves = 64x64 tiles
  const int i0 = (wave >> 6) * 32;
  const int j0 = (wave & 63) * 32;

  // 32-bit A 16x4 layout: lane l holds (row = l&15, k = 2*(l>>4) + {0,1})
  const int r    = lane & 15;
  const int koff = (lane >> 4) << 1;

  const float* a0p = Tw + (size_t)(i0 + r)      * D_N + koff;
  const float* a1p = Tw + (size_t)(i0 + 16 + r) * D_N + koff;
  const float* b0p = Au + (size_t)(j0 + r)      * D_N + koff;
  const float* b1p = Au + (size_t)(j0 + 16 + r) * D_N + koff;

  v8f c00 = {}, c01 = {}, c10 = {}, c11 = {};
  for (int d = 0; d < D_N; d += 4) {
    v2f a0 = *(const v2f*)(a0p + d);
    v2f a1 = *(const v2f*)(a1p + d);
    v2f b0 = *(const v2f*)(b0p + d);
    v2f b1 = *(const v2f*)(b1p + d);
    c00 = __builtin_amdgcn_wmma_f32_16x16x4_f32(false, a0, false, b0, (short)0, c00, false, false);
    c01 = __builtin_amdgcn_wmma_f32_16x16x4_f32(false, a0, false, b1, (short)0, c01, false, false);
    c10 = __builtin_amdgcn_wmma_f32_16x16x4_f32(false, a1, false, b0, (short)0, c10, false, false);
    c11 = __builtin_amdgcn_wmma_f32_16x16x4_f32(false, a1, false, b1, (short)0, c11, false, false);
  }

  // C/D layout: VGPR v -> lanes0-15: (M=v, N=lane); lanes16-31: (M=v+8, N=lane-16)
  float lmax = 0.0f;
  const int nlocal  = lane & 15;
  const int rowskew = (lane >> 4) << 3;
  auto emit = [&](const v8f& c, int ig, int jg) {
    int col = jg + nlocal;
    float aj = awv[col];
#pragma unroll
    for (int v = 0; v < 8; ++v) {
      int row = ig + rowskew + v;
      float dist = twv[row] + aj - 2.0f * c[v];
      float dd = sqrtf(fmaxf(dist, 0.0f));
      Mraw[(size_t)row * B_N + col] = dd;
      lmax = fmaxf(lmax, dd);
    }
  };
  emit(c00, i0,      j0);
  emit(c01, i0,      j0 + 16);
  emit(c10, i0 + 16, j0);
  emit(c11, i0 + 16, j0 + 16);

  __shared__ float smax[256];
  smax[tid] = lmax;
  __syncthreads();
  for (int off = 128; off > 0; off >>= 1) {
    if (tid < off) smax[tid] = fmaxf(smax[tid], smax[tid + off]);
    __syncthreads();
  }
  if (tid == 0) atomicMax(maxbits, __float_as_uint(smax[0]));  // nonneg floats
}

// ---------------------------------------------------------------------------
// Kernel 3: K = exp(-Mraw / (eps * max)) in place; capture diag(K)
// ---------------------------------------------------------------------------
__global__ __launch_bounds__(256) void expk_kernel(
    float* __restrict__ Kmat, float* __restrict__ Kd,
    const unsigned int* __restrict__ maxbits) {
  size_t idx = (size_t)blockIdx.x * 256 + threadIdx.x;
  float maxv  = __uint_as_float(*maxbits);
  float scale = 1.0f / (EPS * maxv);
  float k = expf(-Kmat[idx] * scale);
  Kmat[idx] = k;
  int row = (int)(idx >> 11);
  int col = (int)(idx & (B_N - 1));
  if (row == col) Kd[row] = k;
}

// ---------------------------------------------------------------------------
// Sinkhorn half-step A: v = (1/B) / (K^T u)
//   128 blocks = 16 i-chunks x 8 j-blocks. Deterministic fixed-order
//   partial reduction; last-arriving block finalizes (fused divide).
// ---------------------------------------------------------------------------
__global__ __launch_bounds__(256) void colsum_kernel(
    const float* __restrict__ Kmat, const float* __restrict__ u,
    float* __restrict__ vout, float* __restrict__ partial,
    unsigned int* __restrict__ counter) {
  const int tid = threadIdx.x;
  const int c  = blockIdx.x >> 3;     // i-chunk 0..15
  const int jb = blockIdx.x & 7;      // j-block 0..7
  const int j  = jb * 256 + tid;
  const int ibase = c * 128;
  float acc = 0.0f;
#pragma unroll 4
  for (int ii = 0; ii < 128; ++ii)
    acc += Kmat[(size_t)(ibase + ii) * B_N + j] * u[ibase + ii];
  partial[c * B_N + j] = acc;
  __threadfence();
  __syncthreads();
  __shared__ unsigned int arrival;
  if (tid == 0) arrival = atomicAdd(counter, 1u);
  __syncthreads();
  if (arrival == 127u) {               // last block finalizes all columns
    for (int jj = tid; jj < B_N; jj += 256) {
      float s = 0.0f;
#pragma unroll
      for (int cc = 0; cc < 16; ++cc) s += partial[cc * B_N + jj];
      vout[jj] = INV_B / s;
    }
    __syncthreads();
    if (tid == 0) *counter = 0u;       // reset for next launch (stream-ordered)
  }
}

// ---------------------------------------------------------------------------
// Sinkhorn half-step B: u = (1/B) / (K v), one wave32 per row
// ---------------------------------------------------------------------------
__global__ __launch_bounds__(256) void rowsum_kernel(
    const float* __restrict__ Kmat, const float* __restrict__ v,
    float* __restrict__ uout) {
  const int lane = threadIdx.x & 31;
  const int row  = blockIdx.x * 8 + (threadIdx.x >> 5);
  const float* kr = Kmat + (size_t)row * B_N;
  float acc = 0.0f;
#pragma unroll 4
  for (int j = lane; j < B_N; j += 32) acc += kr[j] * v[j];
#pragma unroll
  for (int off = 16; off > 0; off >>= 1) acc += __shfl_xor(acc, off, 32);
  if (lane == 0) uout[row] = INV_B / acc;
}

// ---------------------------------------------------------------------------
// Final: loss = -(1/B) * sum_i log(u[i] * Kdiag[i] * v[i])
// ---------------------------------------------------------------------------
__global__ __launch_bounds__(256) void loss_kernel(
    const float* __restrict__ u, const float* __restrict__ Kd,
    const float* __restrict__ v, float* __restrict__ out) {
  __shared__ float s[256];
  float acc = 0.0f;
  for (int i = threadIdx.x; i < B_N; i += 256)
    acc += logf(u[i] * Kd[i] * v[i]);
  s[threadIdx.x] = acc;
  __syncthreads();
  for (int off = 128; off > 0; off >>= 1) {
    if (threadIdx.x < off) s[threadIdx.x] += s[threadIdx.x + off];
    __syncthreads();
  }
  if (threadIdx.x == 0) out[0] = -s[0] * INV_B;
}

// ---------------------------------------------------------------------------
// Host launcher (graph-capture safe: only stream launches, no alloc/sync)
// ---------------------------------------------------------------------------
extern "C" void kernel_launch(void* const* d_in, const int* in_sizes, int n_in,
                              void* d_out, int out_size, void* d_ws, size_t ws_size,
                              hipStream_t stream) {
  const float* audio = (const float*)d_in[0];   // [2048,1024] f32
  const float* text  = (const float*)d_in[1];   // [2048,1024] f32
  const float* L     = (const float*)d_in[2];   // [1024,1024] f32

  float* ws = (float*)d_ws;
  float* Kmat = ws;                    // 4,194,304 f32  (Mraw, then K in-place)
  float* Tw   = ws + 4194304;          // 2,097,152 f32  (text * w)
  float* w    = ws + 6291456;          // 1024
  float* awv  = ws + 6292480;          // 2048
  float* twv  = ws + 6294528;          // 2048
  float* u    = ws + 6296576;          // 2048
  float* v    = ws + 6298624;          // 2048
  float* Kd   = ws + 6300672;          // 2048
  float* part = ws + 6302720;          // 16*2048
  unsigned int* maxbits = (unsigned int*)(ws + 6335488);
  unsigned int* counter = (unsigned int*)(ws + 6335489);

  prep_kernel<<<2, 1024, 0, stream>>>(L, w, u, maxbits, counter);
  rowprep_kernel<<<B_N, 256, 0, stream>>>(audio, text, w, Tw, awv, twv);
  cross_dist_kernel<<<512, 256, 0, stream>>>(audio, Tw, awv, twv, Kmat, maxbits);
  expk_kernel<<<(B_N * B_N) / 256, 256, 0, stream>>>(Kmat, Kd, maxbits);
  for (int it = 0; it < N_ITER; ++it) {
    colsum_kernel<<<128, 256, 0, stream>>>(Kmat, u, v, part, counter);
    rowsum_kernel<<<256, 256, 0, stream>>>(Kmat, v, u);
  }
  loss_kernel<<<1, 256, 0, stream>>>(u, Kd, v, (float*)d_out);
}